// BitwiseModule_25606595018771
// MI455X (gfx1250) — compile-verified
//
#include <hip/hip_runtime.h>
#include <stdint.h>

// ---------------------------------------------------------------------------
// BitwiseModule on MI455X (gfx1250)
//
// Pure HBM-streaming problem: 512 MB of traffic, ~22us roofline at 23.3 TB/s.
// No GEMM structure -> WMMA does not apply. We use the CDNA5 async
// global<->LDS DMA path (ASYNCcnt) so the bulk copy never touches VGPRs, and
// the tiny per-row fixup (two +1.0f adds) is done directly in LDS.
// ---------------------------------------------------------------------------

#define D_MODEL        512
#define ROWS_PER_BLOCK 8
#define BLOCK_THREADS  256   // 8 waves of 32 (wave32 on gfx1250)

#if __has_builtin(__builtin_amdgcn_global_load_async_to_lds_b128)
#define HAVE_ASYNC_LD 1
#else
#define HAVE_ASYNC_LD 0
#endif
#if __has_builtin(__builtin_amdgcn_global_store_async_from_lds_b128)
#define HAVE_ASYNC_ST 1
#else
#define HAVE_ASYNC_ST 0
#endif

// Builtin param types (from clang diagnostic): pointer to 16B int vector,
// AS1 ("__device__") for the global side, AS3 ("__shared__") for the LDS side.
typedef int v4i __attribute__((ext_vector_type(4)));
typedef __attribute__((address_space(1))) v4i* gptr_b128;
typedef __attribute__((address_space(3))) v4i* lptr_b128;

// Async copy: 16B per lane per instruction; OFF is a byte immediate.
// Generic->AS casts go through integers: low 32 bits of a generic LDS pointer
// are the workgroup-relative LDS byte offset (flat-aperture layout), which is
// exactly what the async instruction's VDST/VSRC VGPR wants.
template <int OFF>
__device__ __forceinline__ void stage_in(const float* g, float* l) {
#if HAVE_ASYNC_LD
  __builtin_amdgcn_global_load_async_to_lds_b128(
      (gptr_b128)(uintptr_t)g, (lptr_b128)(uint32_t)(uintptr_t)l, OFF, 0);
#else
  *(float4*)((char*)l + OFF) = *(const float4*)((const char*)g + OFF);
#endif
}

template <int OFF>
__device__ __forceinline__ void stage_out(float* g, float* l) {
#if HAVE_ASYNC_ST
  __builtin_amdgcn_global_store_async_from_lds_b128(
      (gptr_b128)(uintptr_t)g, (lptr_b128)(uint32_t)(uintptr_t)l, OFF, 0);
#else
  *(float4*)((char*)g + OFF) = *(const float4*)((const char*)l + OFF);
#endif
}

__device__ __forceinline__ void wait_async_0() {
#if __has_builtin(__builtin_amdgcn_s_wait_asynccnt)
  __builtin_amdgcn_s_wait_asynccnt(0);
#else
#if HAVE_ASYNC_LD || HAVE_ASYNC_ST
  asm volatile("s_wait_asynccnt 0" ::: "memory");
#endif
#endif
  asm volatile("" ::: "memory");
}

// Make the wave's own LDS stores visible to the async store engine
// (DS ops and async VMEM are different counter domains -> explicit DScnt wait).
__device__ __forceinline__ void lds_publish_fence() {
#if HAVE_ASYNC_ST
#if __has_builtin(__builtin_amdgcn_s_wait_dscnt)
  __builtin_amdgcn_s_wait_dscnt(0);
#else
  asm volatile("s_wait_dscnt 0" ::: "memory");
#endif
#endif
  asm volatile("" ::: "memory");
}

__global__ __launch_bounds__(BLOCK_THREADS) void
bitwise_rowfix_kernel(const float* __restrict__ x, float* __restrict__ out,
                      int n_rows) {
  __shared__ float sbuf[ROWS_PER_BLOCK * D_MODEL];  // 16 KB / block

  const int lane = threadIdx.x & 31;
  const int wave = threadIdx.x >> 5;
  const int row  = blockIdx.x * ROWS_PER_BLOCK + wave;
  if (row >= n_rows) return;  // no barriers below; early-out is safe

  const size_t rb   = (size_t)row * D_MODEL;
  const float* gin  = x   + rb + lane * 4;   // lane's 16B slice of the row
  float*       gout = out + rb + lane * 4;
  float* lrow = &sbuf[wave * D_MODEL];
  float* lme  = lrow + lane * 4;

  // ---- stream row into LDS: 4 x 512B fully-coalesced async b128 loads ----
  stage_in<0>(gin, lme);
  stage_in<512>(gin, lme);
  stage_in<1024>(gin, lme);
  stage_in<1536>(gin, lme);
  wait_async_0();

  // ---- decode flags (same-address LDS reads broadcast across the wave) ----
  const float f_and = lrow[0];
  const float f_or  = lrow[1];
  const float f_xor = lrow[2];
  const float f_mk  = lrow[3];
  const bool mk      = f_mk  > 0.5f;
  const bool act_and = (f_and > 0.5f) && mk;
  const bool act_or  = (f_or  > 0.5f) && mk;
  const bool act_xor = (f_xor > 0.5f) && mk;
  const bool active  = act_and || act_or || act_xor;

  // ---- four 16-wide argmaxes, two at a time via the lane halves ----------
  // v1: cols 16..47 = [ALU_LO | ALU_HI], v2: cols 48..79 = [CARRY_LO | CARRY_HI]
  float v1 = lrow[16 + lane];
  float v2 = lrow[48 + lane];
  int   i1 = lane & 15;
  int   i2 = lane & 15;
#pragma unroll
  for (int m = 1; m < 16; m <<= 1) {
    float ov = __shfl_xor(v1, m, 16);   // width=16: reduce within each half
    int   oi = __shfl_xor(i1, m, 16);
    if (ov > v1 || (ov == v1 && oi < i1)) { v1 = ov; i1 = oi; }
    ov = __shfl_xor(v2, m, 16);
    oi = __shfl_xor(i2, m, 16);
    if (ov > v2 || (ov == v2 && oi < i2)) { v2 = ov; i2 = oi; }
  }
  // lexicographic (max value, min index) == jnp.argmax first-max semantics
  const int a = __builtin_amdgcn_readlane(i1, 0) |
                (__builtin_amdgcn_readlane(i1, 16) << 4);
  const int b = __builtin_amdgcn_readlane(i2, 0) |
                (__builtin_amdgcn_readlane(i2, 16) << 4);

  // reference priority: xor overrides or overrides and
  const int res = act_xor ? (a ^ b) : act_or ? (a | b) : act_and ? (a & b) : 0;

  // ---- patch the two output one-hots directly in LDS ---------------------
  if (active && lane == 0) {
    lrow[80 + (res & 15)] += 1.0f;   // OUTPUT_LO + r_lo
    lrow[96 + (res >> 4)] += 1.0f;   // OUTPUT_HI + r_hi
  }
  lds_publish_fence();

  // ---- stream patched row back out: 4 x async b128 stores ----------------
  stage_out<0>(gout, lme);
  stage_out<512>(gout, lme);
  stage_out<1024>(gout, lme);
  stage_out<1536>(gout, lme);
  wait_async_0();  // drain before wave retire (S_ENDPGM would also wait-idle)
}

extern "C" void kernel_launch(void* const* d_in, const int* in_sizes, int n_in,
                              void* d_out, int out_size, void* d_ws,
                              size_t ws_size, hipStream_t stream) {
  (void)n_in; (void)out_size; (void)d_ws; (void)ws_size;
  const float* x   = (const float*)d_in[0];
  float*       out = (float*)d_out;
  const int n_rows = in_sizes[0] / D_MODEL;            // 131072
  const int blocks = (n_rows + ROWS_PER_BLOCK - 1) / ROWS_PER_BLOCK;  // 16384
  bitwise_rowfix_kernel<<<blocks, BLOCK_THREADS, 0, stream>>>(x, out, n_rows);
}